// PaiConvDG_15702400434773
// MI455X (gfx1250) — compile-verified
//
#include <hip/hip_runtime.h>
#include <hip/hip_bf16.h>

// ---------------------------------------------------------------------------
// PaiConvDG on MI455X (gfx1250, wave32).
//  K=20 neighbors, M=16 kernels, C=64, nf=128, B=8, P=4096, GROUP=4.
//  Per-point GEMMs mapped to V_WMMA_F32_16X16X4_F32 (fp32-exact vs reference).
// ---------------------------------------------------------------------------

#define KNN    20
#define MK     16
#define CCH    64
#define NF     128
#define BB     8
#define PP     4096
#define WAVES  4          // waves (== points) per block
#define CW_STR 132        // padded stride for 128-col LDS matrices (bank-conflict-free frags)

typedef float v2f __attribute__((ext_vector_type(2)));
typedef float v8f __attribute__((ext_vector_type(8)));

__device__ __forceinline__ v8f wmma_f32_4(v2f a, v2f b, v8f c) {
    return __builtin_amdgcn_wmma_f32_16x16x4_f32(false, a, false, b, (short)0, c,
                                                 false, false);
}

// ---------------------------------------------------------------------------
// Kernel 1: per-point pipeline -> pre-BN output written to d_out in (B,C,P).
// ---------------------------------------------------------------------------
__global__ __launch_bounds__(32 * WAVES)
void paiconv_point_kernel(const float* __restrict__ x,          // (B,3,P)
                          const float* __restrict__ feature,    // (B,C,P)
                          const long long* __restrict__ nidx,   // (B,P,K) int64
                          const float* __restrict__ kernels,    // (3,M)
                          const float* __restrict__ conv_w,     // (C, 2C)
                          const float* __restrict__ conv_b,     // (C,)
                          float* __restrict__ preout)           // (B,C,P) pre-BN
{
    __shared__ float s_convw[CCH * CW_STR];            // 64 x 128 padded to 132
    __shared__ float s_feats[WAVES][NF * KNN];         // 128 x 20 (A of gemm1)
    __shared__ float s_permT[WAVES][MK * KNN];         // 16  x 20 (B of gemm1, N-major)
    __shared__ float s_aggT [WAVES][MK * CW_STR];      // 16  x 128 pad (B of gemm2, N-major)
    __shared__ float s_xrel [WAVES][KNN * 3];
    __shared__ int   s_idx  [WAVES][KNN];
    __shared__ float s_outb [WAVES][CCH];

    const int w    = threadIdx.x >> 5;
    const int lane = threadIdx.x & 31;
    const int pt   = blockIdx.x * WAVES + w;           // 0 .. B*P-1
    const int b    = pt >> 12;                         // P = 4096
    const int p    = pt & (PP - 1);

    // ---- stage conv_w into LDS (shared by all waves in block) -------------
    for (int i = threadIdx.x; i < CCH * NF; i += blockDim.x)
        s_convw[(i >> 7) * CW_STR + (i & 127)] = conv_w[i];

    // ---- phase A: neighbor indices + relative coordinates -----------------
    {
        const int kk  = (lane < KNN) ? lane : (KNN - 1);
        const int idx = (int)nidx[(b * PP + p) * KNN + kk];
        const float* xb = x + b * 3 * PP;
        float x0 = xb[0 * PP + idx];
        float x1 = xb[1 * PP + idx];
        float x2 = xb[2 * PP + idx];
        float c0 = __shfl(x0, 0);
        float c1 = __shfl(x1, 0);
        float c2 = __shfl(x2, 0);
        if (lane < KNN) {
            s_xrel[w][lane * 3 + 0] = x0 - c0;
            s_xrel[w][lane * 3 + 1] = x1 - c1;
            s_xrel[w][lane * 3 + 2] = x2 - c2;
            s_idx [w][lane] = idx;
        }
    }
    __syncthreads();   // conv_w, xrel, idx visible

    // ---- phase B: perm (relu -> topkmax -> normalize), stored N-major -----
    if (lane < MK) {
        const int m = lane;
        const float k0 = kernels[0 * MK + m];
        const float k1 = kernels[1 * MK + m];
        const float k2 = kernels[2 * MK + m];
        float mx = 0.0f;
        #pragma unroll
        for (int k = 0; k < KNN; ++k) {
            float pv = s_xrel[w][k * 3 + 0] * k0 +
                       s_xrel[w][k * 3 + 1] * k1 +
                       s_xrel[w][k * 3 + 2] * k2;
            if (k == 0 && m == 0) pv += 1.0f;
            pv = fmaxf(pv, 0.0f);
            s_permT[w][m * KNN + k] = pv;
            mx = fmaxf(mx, pv);
        }
        const float thr = 0.5f * mx;
        float s = 0.0f;
        #pragma unroll
        for (int k = 0; k < KNN; ++k) {
            float pv = s_permT[w][m * KNN + k];
            pv = (pv >= thr) ? pv : 0.0f;
            s_permT[w][m * KNN + k] = pv;
            s += pv;
        }
        const float inv = 1.0f / (s + 1e-6f);
        #pragma unroll
        for (int k = 0; k < KNN; ++k)
            s_permT[w][m * KNN + k] *= inv;
    }

    // ---- phase C: build feats (128 x 20) with GROUP=4 interleave ----------
    {
        const float* fb = feature + b * CCH * PP;
        for (int c = lane; c < CCH; c += 32) {
            const float frep = fb[c * PP + s_idx[w][0]];
            const int r0 = ((c & 31) << 2) + (c >> 5);      // broadcast row
            const int r1 = r0 + 2;                          // difference row
            #pragma unroll 4
            for (int k = 0; k < KNN; ++k) {
                float fv = fb[c * PP + s_idx[w][k]];
                s_feats[w][r0 * KNN + k] = frep;
                s_feats[w][r1 * KNN + k] = fv - frep;
            }
        }
    }
    __syncthreads();   // permT, feats visible

    const int Lm   = lane & 15;
    const int half = lane >> 4;        // 0: lanes 0-15, 1: lanes 16-31
    const int koff = half * 2;         // A/B fragment K offset per ISA layout

    // ---- phase D: agg = feats(128x20) @ perm(20x16), D stored transposed --
    for (int t = 0; t < 8; ++t) {
        v8f acc = {0.f, 0.f, 0.f, 0.f, 0.f, 0.f, 0.f, 0.f};
        #pragma unroll
        for (int q = 0; q < 5; ++q) {
            const int k0 = q * 4 + koff;
            v2f a  = *(const v2f*)&s_feats[w][(t * 16 + Lm) * KNN + k0];
            v2f bf = *(const v2f*)&s_permT[w][Lm * KNN + k0];
            acc = wmma_f32_4(a, bf, acc);
        }
        // D frag: lane holds agg[16t + 8*half + r][Lm], r=0..7 -> aggT[n][m]
        float* dst = &s_aggT[w][Lm * CW_STR + t * 16 + half * 8];
        *(float4*)(dst + 0) = make_float4(acc[0], acc[1], acc[2], acc[3]);
        *(float4*)(dst + 4) = make_float4(acc[4], acc[5], acc[6], acc[7]);
    }
    __syncthreads();   // aggT visible

    // ---- phase E: out = conv_w(64x128) @ agg(128x16); max over 16 cols ----
    for (int t = 0; t < 4; ++t) {
        v8f acc = {0.f, 0.f, 0.f, 0.f, 0.f, 0.f, 0.f, 0.f};
        #pragma unroll
        for (int q = 0; q < 32; ++q) {
            const int k0 = q * 4 + koff;
            v2f a  = *(const v2f*)&s_convw[(t * 16 + Lm) * CW_STR + k0];
            v2f bf = *(const v2f*)&s_aggT[w][Lm * CW_STR + k0];
            acc = wmma_f32_4(a, bf, acc);
        }
        // row m = 16t + 8*half + r striped over one 16-lane half -> butterfly max
        #pragma unroll
        for (int r = 0; r < 8; ++r) {
            float v = acc[r];
            v = fmaxf(v, __shfl_xor(v, 1));
            v = fmaxf(v, __shfl_xor(v, 2));
            v = fmaxf(v, __shfl_xor(v, 4));
            v = fmaxf(v, __shfl_xor(v, 8));
            const int row = t * 16 + half * 8 + r;
            if (Lm == 0) s_outb[w][row] = v + conv_b[row];
        }
    }
    __syncthreads();   // outb visible

    // ---- phase F: write pre-BN result, layout (B,C,P) ---------------------
    {
        const int c = lane * 2;
        float* ob = preout + b * CCH * PP + p;
        ob[(c + 0) * PP] = s_outb[w][c + 0];
        ob[(c + 1) * PP] = s_outb[w][c + 1];
    }
}

// ---------------------------------------------------------------------------
// Kernel 2: deterministic per-channel BN statistics (64 blocks, fixed-order
// tree reduction -> scale/shift into d_ws).
// ---------------------------------------------------------------------------
__global__ __launch_bounds__(256)
void bn_stats_kernel(const float* __restrict__ preout,  // (B,C,P)
                     const float* __restrict__ gamma,
                     const float* __restrict__ beta,
                     float* __restrict__ stats)         // [0:64) scale, [64:128) shift
{
    const int c   = blockIdx.x;
    const int tid = threadIdx.x;
    const int N   = BB * PP;

    float sum = 0.0f, sq = 0.0f;
    for (int i = tid; i < N; i += 256) {
        const int b = i >> 12;
        const int p = i & (PP - 1);
        float v = preout[(b * CCH + c) * PP + p];
        sum += v;
        sq  += v * v;
    }
    __shared__ float ssum[256];
    __shared__ float ssq[256];
    ssum[tid] = sum; ssq[tid] = sq;
    __syncthreads();
    for (int s = 128; s > 0; s >>= 1) {
        if (tid < s) { ssum[tid] += ssum[tid + s]; ssq[tid] += ssq[tid + s]; }
        __syncthreads();
    }
    if (tid == 0) {
        const float mean = ssum[0] / (float)N;
        const float var  = ssq[0] / (float)N - mean * mean;
        const float sc   = gamma[c] * rsqrtf(var + 1e-5f);
        stats[c]       = sc;
        stats[64 + c]  = beta[c] - mean * sc;
    }
}

// ---------------------------------------------------------------------------
// Kernel 3: apply BN + residual, in place on d_out.
// ---------------------------------------------------------------------------
__global__ __launch_bounds__(256)
void bn_apply_kernel(float* __restrict__ out,           // (B,C,P), pre-BN in
                     const float* __restrict__ feature, // (B,C,P)
                     const float* __restrict__ stats)
{
    const int idx = blockIdx.x * 256 + threadIdx.x;
    const int c   = (idx >> 12) & (CCH - 1);
    out[idx] = out[idx] * stats[c] + stats[64 + c] + feature[idx];
}

// ---------------------------------------------------------------------------
extern "C" void kernel_launch(void* const* d_in, const int* in_sizes, int n_in,
                              void* d_out, int out_size, void* d_ws, size_t ws_size,
                              hipStream_t stream) {
    (void)in_sizes; (void)n_in; (void)out_size; (void)ws_size;

    const float*     x        = (const float*)d_in[0];
    const float*     feature  = (const float*)d_in[1];
    const long long* nidx     = (const long long*)d_in[2];   // int64 indices
    const float*     kernels  = (const float*)d_in[3];
    const float*     conv_w   = (const float*)d_in[4];
    const float*     conv_b   = (const float*)d_in[5];
    const float*     gamma    = (const float*)d_in[6];
    const float*     beta     = (const float*)d_in[7];

    float* out   = (float*)d_out;
    float* stats = (float*)d_ws;    // 128 floats

    const int npts = BB * PP;                     // 32768 points
    paiconv_point_kernel<<<npts / WAVES, 32 * WAVES, 0, stream>>>(
        x, feature, nidx, kernels, conv_w, conv_b, out);

    bn_stats_kernel<<<CCH, 256, 0, stream>>>(out, gamma, beta, stats);

    const int total = BB * CCH * PP;              // 2M elements
    bn_apply_kernel<<<total / 256, 256, 0, stream>>>(out, feature, stats);
}